// SAGE_60851096649749
// MI455X (gfx1250) — compile-verified
//
#include <hip/hip_runtime.h>
#include <math.h>

#define N_NODES 100000
#define N_EDGES 1600000
#define DF      128          // feature width of x/h0/h1
#define D_OUT   64
#define BN_EPS  1e-5f

typedef __attribute__((ext_vector_type(2))) float v2f;
typedef __attribute__((ext_vector_type(8))) float v8f;

// ---------------------------------------------------------------- CSR build
__global__ void k_zero_i32(int* __restrict__ p, int n) {
    int i = blockIdx.x * blockDim.x + threadIdx.x;
    if (i < n) p[i] = 0;
}

__global__ void k_count(const int* __restrict__ dst, int* __restrict__ cnt, int e) {
    int i = blockIdx.x * blockDim.x + threadIdx.x;
    if (i < e) atomicAdd(&cnt[dst[i]], 1);
}

// Single-block chunked exclusive scan: offs/cursor from cnt; also emits invdeg.
__global__ __launch_bounds__(1024) void k_scan_invdeg(
    const int* __restrict__ cnt, int* __restrict__ offs, int* __restrict__ cursor,
    float* __restrict__ invd, int n)
{
    __shared__ int smem[1024];
    __shared__ int running;
    if (threadIdx.x == 0) running = 0;
    __syncthreads();
    for (int base = 0; base < n; base += 1024) {
        int i = base + (int)threadIdx.x;
        int v = (i < n) ? cnt[i] : 0;
        smem[threadIdx.x] = v;
        __syncthreads();
        #pragma unroll
        for (int off = 1; off < 1024; off <<= 1) {
            int t = (threadIdx.x >= (unsigned)off) ? smem[threadIdx.x - off] : 0;
            __syncthreads();
            smem[threadIdx.x] += t;
            __syncthreads();
        }
        int excl = smem[threadIdx.x] - v;
        int r = running;                       // last updated before prior barrier
        if (i < n) {
            offs[i]   = r + excl;
            cursor[i] = r + excl;
            invd[i]   = 1.0f / fmaxf((float)v, 1.0f);
        }
        __syncthreads();                       // all reads of running done
        if (threadIdx.x == 0) running += smem[1023];
        __syncthreads();
    }
    if (threadIdx.x == 0) offs[n] = running;   // == E
}

__global__ void k_fill(const int* __restrict__ src, const int* __restrict__ dst,
                       int* __restrict__ cursor, int* __restrict__ adj, int e) {
    int i = blockIdx.x * blockDim.x + threadIdx.x;
    if (i < e) {
        int pos = atomicAdd(&cursor[dst[i]], 1);
        adj[pos] = src[i];
    }
}

// ------------------------------------------ gather-mean (no atomics, coalesced)
// One 128-thread block per destination node; sums neighbor rows, scales by 1/deg.
__global__ __launch_bounds__(DF) void k_gather_mean(
    const float* __restrict__ x, const int* __restrict__ offs,
    const int* __restrict__ adj, const float* __restrict__ invd,
    float* __restrict__ agg)
{
    int node = blockIdx.x;
    int f = threadIdx.x;
    int s = offs[node], e = offs[node + 1];
    float acc = 0.0f;
    for (int j = s; j < e; ++j)
        acc += x[(long)adj[j] * DF + f];
    agg[(long)node * DF + f] = acc * invd[node];
}

// ------------------------------------------------- fused SAGE GEMM (WMMA f32)
// out[16x16 tile] = agg @ Wl + bias + xin @ Wr   [+ BN + ReLU]
// One wave per tile, V_WMMA_F32_16X16X4_F32 over K=128 (agg pre-scaled by 1/deg).
__global__ __launch_bounds__(128) void k_sage_gemm(
    const float* __restrict__ agg, const float* __restrict__ xin,
    const float* __restrict__ Wl, const float* __restrict__ bl,
    const float* __restrict__ Wr,
    const float* __restrict__ g, const float* __restrict__ bb,
    const float* __restrict__ m, const float* __restrict__ v,
    float* __restrict__ out, int ncols, int use_bn)
{
    const int K = DF;
    int lane = threadIdx.x;                       // 0..31 (wave32)
    int tile = blockIdx.x * 4 + threadIdx.y;      // 4 waves / block
    int tilesPerRow = ncols >> 4;
    int tm = tile / tilesPerRow;
    int tn = tile - tm * tilesPerRow;
    if (tm * 16 >= N_NODES) return;               // wave-uniform

    int half = lane >> 4;                         // A/B K-half select
    int l16  = lane & 15;
    int arow = tm * 16 + l16;                     // A rows (halves alias rows)
    int bcol = tn * 16 + l16;                     // B/D columns
    const float* aP = agg + (long)arow * K;
    const float* xP = xin + (long)arow * K;

    v8f acc = {};
    #pragma unroll 8
    for (int k = 0; k < K; k += 4) {              // agg @ Wl
        int ka = k + half * 2;
        float2 av = *(const float2*)(aP + ka);
        v2f a, b;
        a.x = av.x;  a.y = av.y;
        b.x = Wl[ka * ncols + bcol];
        b.y = Wl[(ka + 1) * ncols + bcol];
        acc = __builtin_amdgcn_wmma_f32_16x16x4_f32(false, a, false, b,
                                                    (short)0, acc, false, false);
    }
    #pragma unroll 8
    for (int k = 0; k < K; k += 4) {              // xin @ Wr
        int ka = k + half * 2;
        float2 xv = *(const float2*)(xP + ka);
        v2f a, b;
        a.x = xv.x;  a.y = xv.y;
        b.x = Wr[ka * ncols + bcol];
        b.y = Wr[(ka + 1) * ncols + bcol];
        acc = __builtin_amdgcn_wmma_f32_16x16x4_f32(false, a, false, b,
                                                    (short)0, acc, false, false);
    }

    // epilogue: D layout — VGPR i: lanes0-15 -> M=i, lanes16-31 -> M=i+8
    float bias = bl[bcol];
    float gg = 0.f, bv = 0.f, mm = 0.f, isd = 0.f;
    if (use_bn) {
        gg = g[bcol]; bv = bb[bcol]; mm = m[bcol];
        isd = rsqrtf(v[bcol] + BN_EPS);
    }
    #pragma unroll
    for (int i2 = 0; i2 < 8; ++i2) {
        int row = tm * 16 + i2 + half * 8;
        float y = acc[i2] + bias;
        if (use_bn) y = fmaxf(gg * (y - mm) * isd + bv, 0.0f);
        out[(long)row * ncols + bcol] = y;
    }
}

// ------------------------------------------------- in-place log_softmax (64)
__global__ void k_logsoftmax(float* __restrict__ h, int n) {
    int lane = threadIdx.x;                       // 0..31
    int row  = blockIdx.x * blockDim.y + threadIdx.y;
    if (row >= n) return;
    float* p = h + (long)row * D_OUT;
    float a = p[lane], c = p[lane + 32];
    float mx = fmaxf(a, c);
    #pragma unroll
    for (int off = 16; off >= 1; off >>= 1)
        mx = fmaxf(mx, __shfl_xor(mx, off, 32));
    float s = __expf(a - mx) + __expf(c - mx);
    #pragma unroll
    for (int off = 16; off >= 1; off >>= 1)
        s += __shfl_xor(s, off, 32);
    float l = __logf(s);
    p[lane]      = a - mx - l;
    p[lane + 32] = c - mx - l;
}

// ---------------------------------------------------------------------------
extern "C" void kernel_launch(void* const* d_in, const int* in_sizes, int n_in,
                              void* d_out, int out_size, void* d_ws, size_t ws_size,
                              hipStream_t stream) {
    const float* x   = (const float*)d_in[0];
    const int*   ei  = (const int*)d_in[1];
    const int*   src = ei;
    const int*   dst = ei + N_EDGES;
    const float *Wl0 = (const float*)d_in[2],  *bl0 = (const float*)d_in[3],
                *Wr0 = (const float*)d_in[4],  *g0  = (const float*)d_in[5],
                *b0  = (const float*)d_in[6],  *m0  = (const float*)d_in[7],
                *v0  = (const float*)d_in[8],
                *Wl1 = (const float*)d_in[9],  *bl1 = (const float*)d_in[10],
                *Wr1 = (const float*)d_in[11], *g1  = (const float*)d_in[12],
                *b1  = (const float*)d_in[13], *m1  = (const float*)d_in[14],
                *v1  = (const float*)d_in[15],
                *Wl2 = (const float*)d_in[16], *bl2 = (const float*)d_in[17],
                *Wr2 = (const float*)d_in[18];
    float* out = (float*)d_out;

    // workspace layout (ints first, then floats)
    int*   iws    = (int*)d_ws;
    int*   cnt    = iws;                            // N
    int*   offs   = cnt + N_NODES;                  // N+1
    int*   cursor = offs + N_NODES + 1;             // N
    int*   adj    = cursor + N_NODES;               // E
    float* fws    = (float*)(adj + N_EDGES);
    float* invd   = fws;                            // N
    float* agg    = invd + N_NODES;                 // N*128
    float* h0     = agg + (long)N_NODES * DF;       // N*128
    float* h1     = h0  + (long)N_NODES * DF;       // N*128

    const dim3 gemmBlk(32, 4);
    const int  tiles128 = (N_NODES / 16) * (DF / 16);     // 50000
    const int  tiles64  = (N_NODES / 16) * (D_OUT / 16);  // 25000
    const int  eBlocks  = (N_EDGES + 255) / 256;

    // ---- CSR build (once per call) + inverse degrees
    k_zero_i32<<<(N_NODES + 255) / 256, 256, 0, stream>>>(cnt, N_NODES);
    k_count<<<eBlocks, 256, 0, stream>>>(dst, cnt, N_EDGES);
    k_scan_invdeg<<<1, 1024, 0, stream>>>(cnt, offs, cursor, invd, N_NODES);
    k_fill<<<eBlocks, 256, 0, stream>>>(src, dst, cursor, adj, N_EDGES);

    // ---- layer 0: x -> h0
    k_gather_mean<<<N_NODES, DF, 0, stream>>>(x, offs, adj, invd, agg);
    k_sage_gemm<<<tiles128 / 4, gemmBlk, 0, stream>>>(
        agg, x, Wl0, bl0, Wr0, g0, b0, m0, v0, h0, DF, 1);

    // ---- layer 1: h0 -> h1
    k_gather_mean<<<N_NODES, DF, 0, stream>>>(h0, offs, adj, invd, agg);
    k_sage_gemm<<<tiles128 / 4, gemmBlk, 0, stream>>>(
        agg, h0, Wl1, bl1, Wr1, g1, b1, m1, v1, h1, DF, 1);

    // ---- layer 2: h1 -> logits in d_out, then in-place log_softmax
    k_gather_mean<<<N_NODES, DF, 0, stream>>>(h1, offs, adj, invd, agg);
    k_sage_gemm<<<tiles64 / 4, gemmBlk, 0, stream>>>(
        agg, h1, Wl2, bl2, Wr2, nullptr, nullptr, nullptr, nullptr,
        out, D_OUT, 0);

    k_logsoftmax<<<N_NODES / 8, dim3(32, 8), 0, stream>>>(out, N_NODES);
}